// GNN_41446434406489
// MI455X (gfx1250) — compile-verified
//
#include <hip/hip_runtime.h>
#include <hip/hip_bf16.h>

#define N_NODES 1500
#define NEDGE   48000
#define FIN     128
#define FOUT    32
#define NHEAD   4
#define EDIM    16
#define ETOT    (NEDGE + N_NODES)
#define HC      (NHEAD * FOUT)       // 128

typedef __attribute__((ext_vector_type(2))) float v2f;
typedef __attribute__((ext_vector_type(8))) float v8f;

// ---------- ordered-float encoding for atomic segment-max ----------
static __device__ __forceinline__ unsigned encf(float f) {
    unsigned u = __float_as_uint(f);
    unsigned mask = (u >> 31) ? 0xFFFFFFFFu : 0x80000000u;
    return u ^ mask;
}
static __device__ __forceinline__ float decf(unsigned u) {
    unsigned mask = (u >> 31) ? 0x80000000u : 0xFFFFFFFFu;
    return __uint_as_float(u ^ mask);
}
#define ENC_NEG_INF 0x007FFFFFu  // encf(-inf)

// ---------- fills ----------
__global__ void gnn_fill_f32(float* p, float v, int n) {
    int i = blockIdx.x * blockDim.x + threadIdx.x;
    if (i < n) p[i] = v;
}
__global__ void gnn_fill_u32(unsigned* p, unsigned v, int n) {
    int i = blockIdx.x * blockDim.x + threadIdx.x;
    if (i < n) p[i] = v;
}

// ---------- tiny precompute: w_eff[h][d] = sum_c We[d, h*C+c]*att_e[h,c]; probs ----------
__global__ void gnn_weff_probs(const float* __restrict__ We, const float* __restrict__ att_e,
                               const float* __restrict__ fc1_b, const float* __restrict__ fc2_w,
                               const float* __restrict__ fc2_b,
                               float* __restrict__ weff, float* __restrict__ probs) {
    int tid = threadIdx.x;
    if (tid < EDIM * NHEAD) {
        int d = tid % EDIM, hd = tid / EDIM;
        float v = 0.f;
        for (int c = 0; c < FOUT; ++c)
            v += We[d * HC + hd * FOUT + c] * att_e[hd * FOUT + c];
        weff[hd * EDIM + d] = v;
    }
    if (tid == 0) {
        float l0 = fc2_b[0], l1 = fc2_b[1];
        for (int j = 0; j < 32; ++j) {
            float r = fmaxf(fc1_b[j], 0.f);
            l0 += r * fc2_w[j * 2 + 0];
            l1 += r * fc2_w[j * 2 + 1];
        }
        float mx = fmaxf(l0, l1);
        float e0 = __expf(l0 - mx), e1 = __expf(l1 - mx);
        probs[0] = e0 / (e0 + e1);
        probs[1] = e1 / (e0 + e1);
    }
}

// ---------- in-degree + summed edge attrs for self-loop fill_value='mean' ----------
__global__ void gnn_degree(const int* __restrict__ dst0, const float* __restrict__ ef,
                           float* __restrict__ cnt, float* __restrict__ lattr) {
    int e = blockIdx.x * blockDim.x + threadIdx.x;
    if (e >= NEDGE) return;
    int d = dst0[e];
    atomicAdd(&cnt[d], 1.0f);
    for (int k = 0; k < EDIM; ++k)
        atomicAdd(&lattr[d * EDIM + k], ef[e * EDIM + k]);
}
__global__ void gnn_loopattr_div(const float* __restrict__ cnt, float* __restrict__ lattr) {
    int i = blockIdx.x * blockDim.x + threadIdx.x;
    if (i >= N_NODES * EDIM) return;
    float c = cnt[i / EDIM];
    lattr[i] /= fmaxf(c, 1.0f);
}

// ---------- WMMA f32 GEMM: C[M,N] = A[M,K] @ B[K,N], row-major, K%4==0, N%16==0 ----------
__global__ void gnn_wmma_gemm_f32(const float* __restrict__ A, const float* __restrict__ B,
                                  float* __restrict__ C, int M, int N, int K) {
    int tileM = blockIdx.x * 16, tileN = blockIdx.y * 16;
    int lane = threadIdx.x;
    int half = lane >> 4, l16 = lane & 15;
    int row = tileM + l16;
    if (row >= M) row = M - 1;           // clamp (no divergence; store is guarded)
    int col = tileN + l16;
    v8f acc = {};
    for (int k0 = 0; k0 < K; k0 += 4) {
        int ka = k0 + 2 * half;
        v2f a, b;
        a.x = A[row * K + ka];
        a.y = A[row * K + ka + 1];
        b.x = B[ka * N + col];
        b.y = B[(ka + 1) * N + col];
        acc = __builtin_amdgcn_wmma_f32_16x16x4_f32(false, a, false, b,
                                                    (short)0, acc, false, false);
    }
    for (int r = 0; r < 8; ++r) {
        int orow = tileM + r + 8 * half;
        if (orow < M) C[orow * N + col] = acc[r];
    }
}

// ---------- per-node attention dots: s[i,h] = h[i,h,:].a_src[h], t likewise ----------
__global__ void gnn_node_dots(const float* __restrict__ hmat, const float* __restrict__ asrc,
                              const float* __restrict__ adst, float* __restrict__ s,
                              float* __restrict__ t) {
    int idx = blockIdx.x * blockDim.x + threadIdx.x;
    if (idx >= N_NODES * NHEAD) return;
    int i = idx / NHEAD, hd = idx % NHEAD;
    const float* hv = hmat + i * HC + hd * FOUT;
    float sv = 0.f, tv = 0.f;
    for (int c = 0; c < FOUT; ++c) {
        sv += hv[c] * asrc[hd * FOUT + c];
        tv += hv[c] * adst[hd * FOUT + c];
    }
    s[idx] = sv;
    t[idx] = tv;
}

// ---------- per-edge alpha (leaky_relu) + atomic segment-max ----------
__global__ void gnn_edge_alpha(const int* __restrict__ src0, const int* __restrict__ dst0,
                               const float* __restrict__ ef, const float* __restrict__ lattr,
                               const float* __restrict__ weff, const float* __restrict__ s,
                               const float* __restrict__ t, float* __restrict__ ebuf,
                               unsigned* __restrict__ m, int hasExtra) {
    int e = blockIdx.x * blockDim.x + threadIdx.x;
    if (e >= ETOT) return;
    int si, di;
    const float* attr;
    if (e < NEDGE) { si = src0[e]; di = dst0[e]; attr = ef + e * EDIM; }
    else           { si = di = e - NEDGE;        attr = lattr + si * EDIM; }
    for (int hd = 0; hd < NHEAD; ++hd) {
        float a = s[si * NHEAD + hd] + t[di * NHEAD + hd];
        if (hasExtra) {
            float ex = 0.f;
            for (int d = 0; d < EDIM; ++d) ex += attr[d] * weff[hd * EDIM + d];
            a += ex;
        }
        a = (a > 0.f) ? a : 0.2f * a;     // leaky_relu, slope 0.2
        ebuf[e * NHEAD + hd] = a;
        atomicMax(&m[di * NHEAD + hd], encf(a));
    }
}

// ---------- e = exp(alpha - max); atomic segment-sum ----------
__global__ void gnn_edge_expsum(const int* __restrict__ dst0, const float* __restrict__ ebuf_in,
                                const unsigned* __restrict__ m, float* __restrict__ ebuf,
                                float* __restrict__ ssum) {
    int e = blockIdx.x * blockDim.x + threadIdx.x;
    if (e >= ETOT) return;
    int di = (e < NEDGE) ? dst0[e] : (e - NEDGE);
    for (int hd = 0; hd < NHEAD; ++hd) {
        float a = ebuf_in[e * NHEAD + hd];
        float ev = __expf(a - decf(m[di * NHEAD + hd]));
        ebuf[e * NHEAD + hd] = ev;
        atomicAdd(&ssum[di * NHEAD + hd], ev);
    }
}

// ---------- weighted aggregation: accum[dst,h,:] += (e/ssum) * hmat[src,h,:] ----------
__global__ void gnn_aggregate(const int* __restrict__ src0, const int* __restrict__ dst0,
                              const float* __restrict__ ebuf, const float* __restrict__ ssum,
                              const float* __restrict__ hmat, float* __restrict__ accum) {
    int idx = blockIdx.x * blockDim.x + threadIdx.x;
    if (idx >= ETOT * NHEAD) return;
    int e = idx / NHEAD, hd = idx % NHEAD;
    int si, di;
    if (e < NEDGE) { si = src0[e]; di = dst0[e]; }
    else           { si = di = e - NEDGE; }
    float w = ebuf[e * NHEAD + hd] / ssum[di * NHEAD + hd];
    const float* hv = hmat + si * HC + hd * FOUT;
    float* av = accum + di * HC + hd * FOUT;
    for (int c = 0; c < FOUT; ++c) atomicAdd(&av[c], w * hv[c]);
}

// ---------- head mean + bias ----------
__global__ void gnn_head_mean(const float* __restrict__ accum, const float* __restrict__ bias,
                              float* __restrict__ xout) {
    int idx = blockIdx.x * blockDim.x + threadIdx.x;
    if (idx >= N_NODES * FOUT) return;
    int i = idx / FOUT, c = idx % FOUT;
    const float* av = accum + i * HC;
    float v = (av[c] + av[FOUT + c] + av[2 * FOUT + c] + av[3 * FOUT + c]) * 0.25f;
    xout[idx] = v + bias[c];
}

// ---------- constant fill of softmax output (rows >= E are all-zero inputs) ----------
__global__ void gnn_fill_probs(float* __restrict__ out, const float* __restrict__ probs, int nrows) {
    int r = blockIdx.x * blockDim.x + threadIdx.x;
    if (r >= nrows) return;
    out[r * 2 + 0] = probs[0];
    out[r * 2 + 1] = probs[1];
}

// ---------- fc1 via WMMA on gathered pair = [x2[src], x2[dst]] (K=64), relu+bias ----------
__global__ void gnn_fc1_wmma(const float* __restrict__ x2, const int* __restrict__ src0,
                             const int* __restrict__ dst0, const float* __restrict__ fc1w,
                             const float* __restrict__ fc1b, float* __restrict__ hbuf) {
    int tileE = blockIdx.x * 16, tileN = blockIdx.y * 16;
    int lane = threadIdx.x;
    int half = lane >> 4, l16 = lane & 15;
    int er = tileE + l16;
    if (er >= NEDGE) er = NEDGE - 1;
    int es = src0[er], ed = dst0[er];
    int col = tileN + l16;
    v8f acc = {};
    for (int k0 = 0; k0 < 2 * FOUT; k0 += 4) {
        int ka = k0 + 2 * half;           // even pair start, never straddles the 32 boundary
        v2f a, b;
        a.x = (ka < FOUT) ? x2[es * FOUT + ka]     : x2[ed * FOUT + ka - FOUT];
        a.y = (ka + 1 < FOUT) ? x2[es * FOUT + ka + 1] : x2[ed * FOUT + ka + 1 - FOUT];
        b.x = fc1w[ka * 32 + col];
        b.y = fc1w[(ka + 1) * 32 + col];
        acc = __builtin_amdgcn_wmma_f32_16x16x4_f32(false, a, false, b,
                                                    (short)0, acc, false, false);
    }
    for (int r = 0; r < 8; ++r) {
        int orow = tileE + r + 8 * half;
        if (orow < NEDGE) hbuf[orow * 32 + col] = fmaxf(acc[r] + fc1b[col], 0.f);
    }
}

// ---------- fc2 + softmax on the E real edge rows ----------
__global__ void gnn_fc2(const float* __restrict__ hbuf, const float* __restrict__ fc2w,
                        const float* __restrict__ fc2b, float* __restrict__ out) {
    int e = blockIdx.x * blockDim.x + threadIdx.x;
    if (e >= NEDGE) return;
    float l0 = fc2b[0], l1 = fc2b[1];
    const float* hv = hbuf + e * 32;
    for (int j = 0; j < 32; ++j) {
        float r = hv[j];
        l0 += r * fc2w[j * 2 + 0];
        l1 += r * fc2w[j * 2 + 1];
    }
    float mx = fmaxf(l0, l1);
    float e0 = __expf(l0 - mx), e1 = __expf(l1 - mx);
    float inv = 1.0f / (e0 + e1);
    out[e * 2 + 0] = e0 * inv;
    out[e * 2 + 1] = e1 * inv;
}

static inline int cdiv(int a, int b) { return (a + b - 1) / b; }

extern "C" void kernel_launch(void* const* d_in, const int* in_sizes, int n_in,
                              void* d_out, int out_size, void* d_ws, size_t ws_size,
                              hipStream_t stream) {
    const float* x    = (const float*)d_in[0];
    const int*   edges = (const int*)d_in[1];
    const float* ef   = (const float*)d_in[2];
    const float* W1   = (const float*)d_in[3];
    const float* as1  = (const float*)d_in[4];
    const float* ad1  = (const float*)d_in[5];
    const float* We   = (const float*)d_in[6];
    const float* atte = (const float*)d_in[7];
    const float* b1   = (const float*)d_in[8];
    const float* W2   = (const float*)d_in[9];
    const float* as2  = (const float*)d_in[10];
    const float* ad2  = (const float*)d_in[11];
    const float* b2   = (const float*)d_in[12];
    const float* fc1w = (const float*)d_in[13];
    const float* fc1b = (const float*)d_in[14];
    const float* fc2w = (const float*)d_in[15];
    const float* fc2b = (const float*)d_in[16];
    float* out = (float*)d_out;

    const int* src0 = edges;           // edges[0,:]
    const int* dst0 = edges + NEDGE;   // edges[1,:]

    // --- workspace carve-up (floats, 64-elt aligned blocks) ---
    float* ws = (float*)d_ws;
    size_t off = 0;
    auto take = [&](size_t n) { float* p = ws + off; off += (n + 63) & ~size_t(63); return p; };
    float*    h     = take(N_NODES * HC);        // h1 then h2 (reused)
    float*    sbuf  = take(N_NODES * NHEAD);
    float*    tbuf  = take(N_NODES * NHEAD);
    float*    cnt   = take(N_NODES);
    float*    lattr = take(N_NODES * EDIM);
    float*    ebuf  = take((size_t)ETOT * NHEAD);
    unsigned* m     = (unsigned*)take(N_NODES * NHEAD);
    float*    ssum  = take(N_NODES * NHEAD);
    float*    accum = take(N_NODES * HC);
    float*    x1v   = take(N_NODES * FOUT);
    float*    x2v   = take(N_NODES * FOUT);
    float*    weff  = take(NHEAD * EDIM);
    float*    probs = take(2);
    float*    hbuf  = take((size_t)NEDGE * 32);
    (void)ws_size; (void)in_sizes; (void)n_in; (void)out_size;

    const int T = 256;

    // precompute effective edge-attention weights + constant output row
    gnn_weff_probs<<<1, 64, 0, stream>>>(We, atte, fc1b, fc2w, fc2b, weff, probs);

    // self-loop attrs (fill_value='mean')
    gnn_fill_f32<<<cdiv(N_NODES, T), T, 0, stream>>>(cnt, 0.f, N_NODES);
    gnn_fill_f32<<<cdiv(N_NODES * EDIM, T), T, 0, stream>>>(lattr, 0.f, N_NODES * EDIM);
    gnn_degree<<<cdiv(NEDGE, T), T, 0, stream>>>(dst0, ef, cnt, lattr);
    gnn_loopattr_div<<<cdiv(N_NODES * EDIM, T), T, 0, stream>>>(cnt, lattr);

    // ===== layer 1 =====
    gnn_wmma_gemm_f32<<<dim3(cdiv(N_NODES, 16), HC / 16), 32, 0, stream>>>(x, W1, h,
                                                                           N_NODES, HC, FIN);
    gnn_node_dots<<<cdiv(N_NODES * NHEAD, T), T, 0, stream>>>(h, as1, ad1, sbuf, tbuf);
    gnn_fill_u32<<<cdiv(N_NODES * NHEAD, T), T, 0, stream>>>(m, ENC_NEG_INF, N_NODES * NHEAD);
    gnn_fill_f32<<<cdiv(N_NODES * NHEAD, T), T, 0, stream>>>(ssum, 0.f, N_NODES * NHEAD);
    gnn_fill_f32<<<cdiv(N_NODES * HC, T), T, 0, stream>>>(accum, 0.f, N_NODES * HC);
    gnn_edge_alpha<<<cdiv(ETOT, T), T, 0, stream>>>(src0, dst0, ef, lattr, weff,
                                                    sbuf, tbuf, ebuf, m, /*hasExtra=*/1);
    gnn_edge_expsum<<<cdiv(ETOT, T), T, 0, stream>>>(dst0, ebuf, m, ebuf, ssum);
    gnn_aggregate<<<cdiv(ETOT * NHEAD, T), T, 0, stream>>>(src0, dst0, ebuf, ssum, h, accum);
    gnn_head_mean<<<cdiv(N_NODES * FOUT, T), T, 0, stream>>>(accum, b1, x1v);

    // ===== layer 2 =====
    gnn_wmma_gemm_f32<<<dim3(cdiv(N_NODES, 16), HC / 16), 32, 0, stream>>>(x1v, W2, h,
                                                                           N_NODES, HC, FOUT);
    gnn_node_dots<<<cdiv(N_NODES * NHEAD, T), T, 0, stream>>>(h, as2, ad2, sbuf, tbuf);
    gnn_fill_u32<<<cdiv(N_NODES * NHEAD, T), T, 0, stream>>>(m, ENC_NEG_INF, N_NODES * NHEAD);
    gnn_fill_f32<<<cdiv(N_NODES * NHEAD, T), T, 0, stream>>>(ssum, 0.f, N_NODES * NHEAD);
    gnn_fill_f32<<<cdiv(N_NODES * HC, T), T, 0, stream>>>(accum, 0.f, N_NODES * HC);
    gnn_edge_alpha<<<cdiv(ETOT, T), T, 0, stream>>>(src0, dst0, ef, lattr, weff,
                                                    sbuf, tbuf, ebuf, m, /*hasExtra=*/0);
    gnn_edge_expsum<<<cdiv(ETOT, T), T, 0, stream>>>(dst0, ebuf, m, ebuf, ssum);
    gnn_aggregate<<<cdiv(ETOT * NHEAD, T), T, 0, stream>>>(src0, dst0, ebuf, ssum, h, accum);
    gnn_head_mean<<<cdiv(N_NODES * FOUT, T), T, 0, stream>>>(accum, b2, x2v);

    // ===== edge classifier =====
    gnn_fill_probs<<<cdiv(N_NODES * N_NODES, T), T, 0, stream>>>(out, probs,
                                                                 N_NODES * N_NODES);
    gnn_fc1_wmma<<<dim3(cdiv(NEDGE, 16), 2), 32, 0, stream>>>(x2v, src0, dst0,
                                                              fc1w, fc1b, hbuf);
    gnn_fc2<<<cdiv(NEDGE, T), T, 0, stream>>>(hbuf, fc2w, fc2b, out);
}